// CDWCELoss_26173530702140
// MI455X (gfx1250) — compile-verified
//
#include <hip/hip_runtime.h>
#include <hip/hip_bf16.h>

typedef __attribute__((ext_vector_type(2))) float v2f;
typedef __attribute__((ext_vector_type(8))) float v8f;

// Full 32-lane sum using V_WMMA_F32_16X16X4_F32 with a ones B-matrix.
// x goes in A vgpr0 (K=0 for lanes 0-15, K=2 for lanes 16-31), zeros in vgpr1,
// so D[m][n] = x[m] + x[m+16]. Each lane sums its 8 D vgprs (half-sum), then
// one ds_swizzle SWAPX16 completes the 32-lane sum in every lane.
__device__ __forceinline__ float wave_reduce_sum_wmma(float x) {
    v2f a; a[0] = x;    a[1] = 0.0f;
    v2f b; b[0] = 1.0f; b[1] = 1.0f;
    v8f c = {};
    v8f d = __builtin_amdgcn_wmma_f32_16x16x4_f32(
        /*neg_a=*/false, a, /*neg_b=*/false, b,
        /*c_mod=*/(short)0, c, /*reuse_a=*/false, /*reuse_b=*/false);
    float h = ((d[0] + d[1]) + (d[2] + d[3])) + ((d[4] + d[5]) + (d[6] + d[7]));
    h += __int_as_float(__builtin_amdgcn_ds_swizzle(__float_as_int(h), 0x401F)); // xor 16
    return h;
}

#define ROWS_PER_THREAD 4

__global__ __launch_bounds__(256)
void cdwce_partial_kernel(const float* __restrict__ logits,
                          const long long* __restrict__ targets,
                          float* __restrict__ partials,
                          int nrows) {
    const float LOG2E   = 1.4426950408889634f;   // log2(e)
    const float NLN2    = -0.6931471805599453f;  // -ln(2)
    const float ONE_EPS = 1.0f + 1e-7f;

    const long long base =
        ((long long)blockIdx.x * 256 + threadIdx.x) * ROWS_PER_THREAD;

    float acc = 0.0f;
#pragma unroll
    for (int k = 0; k < ROWS_PER_THREAD; ++k) {
        const long long row = base + k;
        if (row < nrows) {
            const float4* p = reinterpret_cast<const float4*>(logits + row * 32);
            __builtin_prefetch(p, 0, 0);   // global_prefetch_b8

            // Load row, pre-scale into log2 domain (softmax is base-invariant).
            float r[32];
#pragma unroll
            for (int i = 0; i < 8; ++i) {
                float4 q = p[i];
                r[4 * i + 0] = q.x * LOG2E; r[4 * i + 1] = q.y * LOG2E;
                r[4 * i + 2] = q.z * LOG2E; r[4 * i + 3] = q.w * LOG2E;
            }

            float m = r[0];
#pragma unroll
            for (int c2 = 1; c2 < 32; ++c2) m = fmaxf(m, r[c2]);

            float s = 0.0f;
#pragma unroll
            for (int c2 = 0; c2 < 32; ++c2) {
                float e = __builtin_amdgcn_exp2f(r[c2] - m);   // raw v_exp_f32
                r[c2] = e;
                s += e;
            }
            const float inv = __builtin_amdgcn_rcpf(s);        // raw v_rcp_f32

            const float tf = (float)targets[row];
            float racc = 0.0f;
#pragma unroll
            for (int c2 = 0; c2 < 32; ++c2) {
                float y  = fmaf(r[c2], -inv, ONE_EPS);         // 1 + eps - p
                float lg = __builtin_amdgcn_logf(y);           // raw v_log_f32 (log2)
                float td = (float)c2 - tf;                     // even power: no fabs
                float d2 = td * td;
                float d6 = d2 * d2 * d2;
                racc = fmaf(lg, d6, racc);                     // sum log2(y)*d^6
            }
            acc += racc;   // -ln2 factored out of the inner loop
        }
    }
    acc *= NLN2;   // convert log2 -> -ln once per thread

    // Re-converged: EXEC all-ones (WMMA requirement).
    float wsum = wave_reduce_sum_wmma(acc);

    __shared__ float wpart[8];
    const int lane = threadIdx.x & 31;
    const int wid  = threadIdx.x >> 5;
    if (lane == 0) wpart[wid] = wsum;
    __syncthreads();

    if (wid == 0) {   // uniform branch: whole wave 0 active, EXEC all-ones
        float v = (lane < 8) ? wpart[lane] : 0.0f;
        v += __int_as_float(__builtin_amdgcn_ds_swizzle(__float_as_int(v), 0x101F)); // xor 4
        v += __int_as_float(__builtin_amdgcn_ds_swizzle(__float_as_int(v), 0x081F)); // xor 2
        v += __int_as_float(__builtin_amdgcn_ds_swizzle(__float_as_int(v), 0x041F)); // xor 1
        if (lane == 0) partials[blockIdx.x] = v;
    }
}

__global__ __launch_bounds__(256)
void cdwce_finalize_kernel(const float* __restrict__ partials, int n,
                           float* __restrict__ out, double invB) {
    __shared__ double sd[256];
    double a = 0.0;
    for (int i = threadIdx.x; i < n; i += 256) a += (double)partials[i];
    sd[threadIdx.x] = a;
    __syncthreads();
    for (int str = 128; str > 0; str >>= 1) {
        if ((int)threadIdx.x < str) sd[threadIdx.x] += sd[threadIdx.x + str];
        __syncthreads();
    }
    if (threadIdx.x == 0) out[0] = (float)(sd[0] * invB);
}

extern "C" void kernel_launch(void* const* d_in, const int* in_sizes, int n_in,
                              void* d_out, int out_size, void* d_ws, size_t ws_size,
                              hipStream_t stream) {
    const float*     logits  = (const float*)d_in[0];
    const long long* targets = (const long long*)d_in[1];   // reference dtype int64

    const int C      = 32;
    const int nrows  = in_sizes[0] / C;                     // 1,048,576
    const int rows_per_block = 256 * ROWS_PER_THREAD;       // 1024
    const int blocks = (nrows + rows_per_block - 1) / rows_per_block;  // 1024

    float* partials = (float*)d_ws;                         // 4 KB of d_ws

    cdwce_partial_kernel<<<blocks, 256, 0, stream>>>(logits, targets, partials, nrows);
    cdwce_finalize_kernel<<<1, 256, 0, stream>>>(partials, blocks, (float*)d_out,
                                                 1.0 / (double)nrows);
}